// GCN_59854664237647
// MI455X (gfx1250) — compile-verified
//
#include <hip/hip_runtime.h>
#include <math.h>

typedef __attribute__((ext_vector_type(2))) float v2f;
typedef __attribute__((ext_vector_type(8))) float v8f;

#define K_IN 128
#define F1 16
#define F2 3

// ---------------- degree / normalization ----------------
__global__ void k_deg_init(float* __restrict__ deg, int n) {
  int i = blockIdx.x * blockDim.x + threadIdx.x;
  if (i < n) deg[i] = 1.0f;  // self-loop
}

__global__ void k_deg_count(const int* __restrict__ dst, float* __restrict__ deg, int E) {
  int e = blockIdx.x * blockDim.x + threadIdx.x;
  if (e < E) atomicAdd(&deg[dst[e]], 1.0f);
}

__global__ void k_dinv(float* __restrict__ deg, int n) {
  int i = blockIdx.x * blockDim.x + threadIdx.x;
  if (i < n) deg[i] = rsqrtf(deg[i]);  // deg >= 1 always (self loops)
}

// ---------------- layer-1 GEMM: h1 = x @ W1 via f32 WMMA ----------------
// One wave per 16-row tile. 16x16 output, K=128 in 32 steps of 4.
__global__ void k_gemm1_wmma(const float* __restrict__ x, const float* __restrict__ W,
                             float* __restrict__ h, int n) {
  int wave = (int)((blockIdx.x * blockDim.x + threadIdx.x) >> 5);
  int lane = threadIdx.x & 31;
  int row0 = wave * 16;
  if (row0 >= n) return;

  if (row0 + 16 <= n) {
    // full tile: WMMA path (wave-uniform branch, EXEC all 1s)
    int m    = lane & 15;        // A: row, B: col, per ISA 16x4 / 4x16 layout
    int koff = (lane >> 4) << 1; // lanes 0-15 -> K+0,K+1 ; lanes 16-31 -> K+2,K+3
    const float* xr = x + (size_t)(row0 + m) * K_IN;
    v8f acc = {};
#pragma unroll
    for (int k = 0; k < K_IN; k += 4) {
      v2f a, b;
      a.x = xr[k + koff];
      a.y = xr[k + koff + 1];
      b.x = W[(k + koff) * F1 + m];
      b.y = W[(k + koff + 1) * F1 + m];
      acc = __builtin_amdgcn_wmma_f32_16x16x4_f32(
          /*neg_a=*/false, a, /*neg_b=*/false, b,
          /*c_mod=*/(short)0, acc, /*reuse_a=*/false, /*reuse_b=*/false);
    }
    // D layout: VGPR r on lane l holds (M = r + 8*(l>>4), N = l&15)
    int half = lane >> 4;
    int col  = lane & 15;
#pragma unroll
    for (int r = 0; r < 8; ++r) {
      int mm = r + 8 * half;
      h[(size_t)(row0 + mm) * F1 + col] = acc[r];
    }
  } else {
    // ragged tail tile (not hit for n=100000): scalar fallback, lanes 0-15
    if (lane < 16) {
      int row = row0 + lane;
      if (row < n) {
        const float* xr = x + (size_t)row * K_IN;
        for (int c = 0; c < F1; ++c) {
          float acc = 0.f;
          for (int k = 0; k < K_IN; ++k) acc += xr[k] * W[k * F1 + c];
          h[(size_t)row * F1 + c] = acc;
        }
      }
    }
  }
}

// ---------------- layer-1 edge aggregation (16 lanes per edge) ----------------
__global__ void k_scatter1(const int* __restrict__ src, const int* __restrict__ dst,
                           const float* __restrict__ h, const float* __restrict__ dinv,
                           float* __restrict__ agg, int E) {
  int gid = blockIdx.x * blockDim.x + threadIdx.x;
  int e = gid >> 4;
  int f = gid & 15;
  if (e >= E) return;
  int s = src[e], d = dst[e];
  float nrm = dinv[s] * dinv[d];
  atomicAdd(&agg[(size_t)d * F1 + f], h[(size_t)s * F1 + f] * nrm);
}

// self-loop contribution + bias + ReLU, in place on agg1
__global__ void k_post1(const float* __restrict__ h, const float* __restrict__ dinv,
                        const float* __restrict__ b, float* __restrict__ agg, int n) {
  int gid = blockIdx.x * blockDim.x + threadIdx.x;
  int i = gid >> 4;
  int f = gid & 15;
  if (i >= n) return;
  float di = dinv[i];
  size_t idx = (size_t)i * F1 + f;
  float v = agg[idx] + h[idx] * di * di + b[f];
  agg[idx] = v > 0.f ? v : 0.f;
}

// ---------------- layer-2 GEMM: h2 = relu_out @ W2 (16 -> 3) ----------------
__global__ void k_gemm2(const float* __restrict__ hr, const float* __restrict__ W2,
                        float* __restrict__ h2, int n) {
  int i = blockIdx.x * blockDim.x + threadIdx.x;
  if (i >= n) return;
  float a0 = 0.f, a1 = 0.f, a2 = 0.f;
#pragma unroll
  for (int f = 0; f < F1; ++f) {
    float v = hr[(size_t)i * F1 + f];
    a0 += v * W2[f * F2 + 0];
    a1 += v * W2[f * F2 + 1];
    a2 += v * W2[f * F2 + 2];
  }
  h2[(size_t)i * F2 + 0] = a0;
  h2[(size_t)i * F2 + 1] = a1;
  h2[(size_t)i * F2 + 2] = a2;
}

// ---------------- layer-2 edge aggregation (1 thread per edge) ----------------
__global__ void k_scatter2(const int* __restrict__ src, const int* __restrict__ dst,
                           const float* __restrict__ h2, const float* __restrict__ dinv,
                           float* __restrict__ agg, int E) {
  int e = blockIdx.x * blockDim.x + threadIdx.x;
  if (e >= E) return;
  int s = src[e], d = dst[e];
  float nrm = dinv[s] * dinv[d];
#pragma unroll
  for (int c = 0; c < F2; ++c)
    atomicAdd(&agg[(size_t)d * F2 + c], h2[(size_t)s * F2 + c] * nrm);
}

// self-loop + bias + log_softmax -> d_out
__global__ void k_final(const float* __restrict__ h2, const float* __restrict__ agg,
                        const float* __restrict__ dinv, const float* __restrict__ b2,
                        float* __restrict__ out, int n) {
  int i = blockIdx.x * blockDim.x + threadIdx.x;
  if (i >= n) return;
  float di = dinv[i];
  float dd = di * di;
  float v0 = agg[(size_t)i * F2 + 0] + h2[(size_t)i * F2 + 0] * dd + b2[0];
  float v1 = agg[(size_t)i * F2 + 1] + h2[(size_t)i * F2 + 1] * dd + b2[1];
  float v2 = agg[(size_t)i * F2 + 2] + h2[(size_t)i * F2 + 2] * dd + b2[2];
  float m = fmaxf(v0, fmaxf(v1, v2));
  float lse = logf(expf(v0 - m) + expf(v1 - m) + expf(v2 - m));
  out[(size_t)i * F2 + 0] = v0 - m - lse;
  out[(size_t)i * F2 + 1] = v1 - m - lse;
  out[(size_t)i * F2 + 2] = v2 - m - lse;
}

extern "C" void kernel_launch(void* const* d_in, const int* in_sizes, int n_in,
                              void* d_out, int out_size, void* d_ws, size_t ws_size,
                              hipStream_t stream) {
  const float* x  = (const float*)d_in[0];
  const int*   ei = (const int*)d_in[1];
  const float* W1 = (const float*)d_in[2];
  const float* b1 = (const float*)d_in[3];
  const float* W2 = (const float*)d_in[4];
  const float* b2 = (const float*)d_in[5];

  const int n = in_sizes[0] / K_IN;   // 100000
  const int E = in_sizes[1] / 2;      // 1600000
  const int* src = ei;
  const int* dst = ei + E;

  // workspace layout (256B-aligned regions)
  auto align = [](size_t v) { return (v + 255) & ~(size_t)255; };
  char* ws = (char*)d_ws;
  size_t o = 0;
  float* dinv = (float*)(ws + o); o += align((size_t)n * sizeof(float));
  float* h1   = (float*)(ws + o); o += align((size_t)n * F1 * sizeof(float));
  float* agg1 = (float*)(ws + o); o += align((size_t)n * F1 * sizeof(float));
  float* h2   = (float*)(ws + o); o += align((size_t)n * F2 * sizeof(float));
  float* agg2 = (float*)(ws + o); o += align((size_t)n * F2 * sizeof(float));
  (void)ws_size; (void)n_in; (void)out_size;

  const int B = 256;

  // zero aggregation buffers (capture-safe)
  hipMemsetAsync(agg1, 0, (size_t)n * F1 * sizeof(float), stream);
  hipMemsetAsync(agg2, 0, (size_t)n * F2 * sizeof(float), stream);

  // degrees -> dinv
  k_deg_init<<<(n + B - 1) / B, B, 0, stream>>>(dinv, n);
  k_deg_count<<<(E + B - 1) / B, B, 0, stream>>>(dst, dinv, E);
  k_dinv<<<(n + B - 1) / B, B, 0, stream>>>(dinv, n);

  // layer 1
  int tiles = (n + 15) / 16;                    // 6250 waves
  int g1 = (tiles * 32 + B - 1) / B;
  k_gemm1_wmma<<<g1, B, 0, stream>>>(x, W1, h1, n);

  long long w1 = (long long)E * F1;
  k_scatter1<<<(int)((w1 + B - 1) / B), B, 0, stream>>>(src, dst, h1, dinv, agg1, E);

  long long p1 = (long long)n * F1;
  k_post1<<<(int)((p1 + B - 1) / B), B, 0, stream>>>(h1, dinv, b1, agg1, n);

  // layer 2
  k_gemm2<<<(n + B - 1) / B, B, 0, stream>>>(agg1, W2, h2, n);
  k_scatter2<<<(E + B - 1) / B, B, 0, stream>>>(src, dst, h2, dinv, agg2, E);
  k_final<<<(n + B - 1) / B, B, 0, stream>>>(h2, agg2, dinv, b2, (float*)d_out, n);
}